// StackAll_75883482186296
// MI455X (gfx1250) — compile-verified
//
#include <hip/hip_runtime.h>

typedef __attribute__((ext_vector_type(2))) float v2f;
typedef __attribute__((ext_vector_type(8))) float v8f;
typedef int v4i __attribute__((vector_size(16)));
typedef __attribute__((address_space(1))) v4i gv4i_t;   // global int4
typedef __attribute__((address_space(3))) v4i lv4i_t;   // LDS int4

#define N_DIM 1024
#define M_DIM 1024
#define D_DIM 512
#define DC 64                          // D-chunk staged in LDS per iteration
#define STRIDE (DC + 4)                // 68 floats: 16B-aligned rows, bank-shifted
#define NCHUNK (D_DIM / DC)

// direct global -> LDS async copy (16B), tracked by ASYNCcnt; no VGPR staging
__device__ __forceinline__ void async_cp16(const float* gsrc, float* ldst) {
    __builtin_amdgcn_global_load_async_to_lds_b128((gv4i_t*)gsrc, (lv4i_t*)ldst,
                                                   /*offset=*/0, /*cpol=*/0);
}
__device__ __forceinline__ void wait_async0() {
    asm volatile("s_wait_asynccnt 0" ::: "memory");
}

// ---------------- row-norm precompute: ws[0..1023]=||z_n||^2, ws[1024..2047]=||zpr_m||^2
__global__ void stackall_norms(const float* __restrict__ z,
                               const float* __restrict__ zpr,
                               float* __restrict__ ws) {
    int gwave = (blockIdx.x * blockDim.x + threadIdx.x) >> 5;
    int lane  = threadIdx.x & 31;
    const float* row = (gwave < N_DIM) ? (z + (size_t)gwave * D_DIM)
                                       : (zpr + (size_t)(gwave - N_DIM) * D_DIM);
    float s = 0.f;
    for (int i = lane; i < D_DIM; i += 32) {
        float v = row[i];
        s = fmaf(v, v, s);
    }
    for (int off = 16; off > 0; off >>= 1) s += __shfl_down(s, off, 32);
    if (lane == 0) ws[gwave] = s;
}

// ---------------- fused L1 / L2 / dot kernel
// block = 256 threads (8 wave32), block tile = 32(n) x 64(m), wave tile = 16x16.
// dp via V_WMMA_F32_16X16X4_F32 (matrix pipe), L1 via VALU from the same LDS tiles,
// l2 = sqrt(||z||^2 + ||zpr||^2 - 2*dp) from precomputed norms.
// Double-buffered LDS fed by GLOBAL_LOAD_ASYNC_TO_LDS_B128 (ASYNCcnt).
__global__ __launch_bounds__(256)
void stackall_main(const float* __restrict__ z, const float* __restrict__ zpr,
                   const float* __restrict__ ws, float* __restrict__ out) {
    __shared__ float zT[2 * 32 * STRIDE];
    __shared__ float pT[2 * 64 * STRIDE];

    const int tid    = threadIdx.x;
    const int lane   = tid & 31;
    const int waveId = tid >> 5;
    const int waveN  = waveId & 1;   // 2 waves along n
    const int waveM  = waveId >> 1;  // 4 waves along m

    const int nBlk = blockIdx.y * 32;
    const int mBlk = blockIdx.x * 64;
    const int n0   = nBlk + waveN * 16;
    const int m0   = mBlk + waveM * 16;

    const int r15 = lane & 15;
    const int hi  = lane >> 4;       // lane-half: K+2/+3 in frags, M+8 in C layout
    const int hiK = hi * 2;

    v8f acc = {};                    // dp accumulators (C layout)
    float4 l1v[8];                   // vector L1 accumulators: no per-iter horizontal
#pragma unroll
    for (int v = 0; v < 8; ++v) l1v[v] = make_float4(0.f, 0.f, 0.f, 0.f);

    // per-thread cooperative-copy coordinates (float4 granularity)
    // z tile: 32 rows x 16 float4 = 512 -> 2 per thread; p tile: 64 x 16 = 1024 -> 4
    auto issue_chunk = [&](int buf, int kb) {
        int idx = tid;
#pragma unroll
        for (int i = 0; i < 2; ++i, idx += 256) {
            int row = idx >> 4, c4 = (idx & 15) << 2;
            async_cp16(z + (size_t)(nBlk + row) * D_DIM + kb + c4,
                       &zT[buf * 32 * STRIDE + row * STRIDE + c4]);
        }
        idx = tid;
#pragma unroll
        for (int i = 0; i < 4; ++i, idx += 256) {
            int row = idx >> 4, c4 = (idx & 15) << 2;
            async_cp16(zpr + (size_t)(mBlk + row) * D_DIM + kb + c4,
                       &pT[buf * 64 * STRIDE + row * STRIDE + c4]);
        }
    };

    issue_chunk(0, 0);
    wait_async0();
    __syncthreads();

    for (int c = 0; c < NCHUNK; ++c) {
        const int buf = c & 1;
        if (c + 1 < NCHUNK) issue_chunk(buf ^ 1, (c + 1) * DC);  // overlap with compute

        const float* zw = &zT[buf * 32 * STRIDE + (waveN * 16) * STRIDE];
        const float* pw = &pT[buf * 64 * STRIDE + (waveM * 16) * STRIDE];

#pragma unroll 4
        for (int k = 0; k < DC; k += 4) {
            float4 pv = *(const float4*)(pw + r15 * STRIDE + k);          // L1 operand
            float2 af = *(const float2*)(zw + r15 * STRIDE + k + hiK);    // A fragment
            float2 bf = *(const float2*)(pw + r15 * STRIDE + k + hiK);    // B fragment
            v2f a, b;
            a.x = af.x;  a.y = af.y;
            b.x = bf.x;  b.y = bf.y;
            acc = __builtin_amdgcn_wmma_f32_16x16x4_f32(false, a, false, b,
                                                        (short)0, acc, false, false);
            // L1: 8 n-rows per lane (half-wave broadcast reads, conflict-free)
#pragma unroll
            for (int v = 0; v < 8; ++v) {
                float4 zv = *(const float4*)(zw + (v + 8 * hi) * STRIDE + k);
                l1v[v].x += fabsf(zv.x - pv.x);
                l1v[v].y += fabsf(zv.y - pv.y);
                l1v[v].z += fabsf(zv.z - pv.z);
                l1v[v].w += fabsf(zv.w - pv.w);
            }
        }

        wait_async0();      // next chunk's LDS writes complete (this thread)
        __syncthreads();    // ... for all threads; and buf fully consumed by all
    }

    // epilogue: out[m][n][{l1,l2,dp}]
    const float* sqz = ws;
    const float* sqp = ws + N_DIM;
    const int m = m0 + r15;
    const float sp = sqp[m];
#pragma unroll
    for (int v = 0; v < 8; ++v) {
        int   n  = n0 + v + 8 * hi;
        float dp = acc[v];
        float l1 = (l1v[v].x + l1v[v].y) + (l1v[v].z + l1v[v].w);
        float l2 = sqrtf(fmaxf(sqz[n] + sp - 2.f * dp, 0.f));
        size_t o = ((size_t)m * N_DIM + n) * 3;
        out[o + 0] = l1;
        out[o + 1] = l2;
        out[o + 2] = dp;
    }
}

extern "C" void kernel_launch(void* const* d_in, const int* in_sizes, int n_in,
                              void* d_out, int out_size, void* d_ws, size_t ws_size,
                              hipStream_t stream) {
    (void)in_sizes; (void)n_in; (void)out_size; (void)ws_size;
    const float* z   = (const float*)d_in[0];
    const float* zpr = (const float*)d_in[1];
    float* out = (float*)d_out;
    float* ws  = (float*)d_ws;   // needs 2048 floats (8 KB) of scratch

    stackall_norms<<<(2 * N_DIM * 32) / 256, 256, 0, stream>>>(z, zpr, ws);
    dim3 grid(M_DIM / 64, N_DIM / 32);
    stackall_main<<<grid, 256, 0, stream>>>(z, zpr, ws, out);
}